// TopkRouter_75058848464994
// MI455X (gfx1250) — compile-verified
//
#include <hip/hip_runtime.h>
#include <hip/hip_bf16.h>

// Problem constants (match reference)
#define B_    4
#define N_    4096
#define D_    1024
#define E_    8
#define TOPK_ 2
#define CAP_  1024
static constexpr size_t BNEC = (size_t)B_ * N_ * E_ * CAP_;   // 134217728 floats per tensor

typedef float f32x2 __attribute__((ext_vector_type(2)));
typedef float f32x4 __attribute__((ext_vector_type(4)));
typedef float f32x8 __attribute__((ext_vector_type(8)));

struct TokRec { int e1; int e2; float p1; float p2; };   // 16 bytes

// ---------------------------------------------------------------------------
// 0) zero the small accumulator block in workspace (ws is poisoned 0xAA)
//    layout (floats): [0] zsum, [1..32] psum[B][E], [33..64] cnt[B][E]
// ---------------------------------------------------------------------------
__global__ void init_ws_kernel(float* __restrict__ accbuf) {
    if (threadIdx.x < 256) accbuf[threadIdx.x] = 0.0f;
}

// ---------------------------------------------------------------------------
// 1) zero-fill dispatch+combine (1 GB) with non-temporal b128 stores
// ---------------------------------------------------------------------------
__global__ void zero_kernel(f32x4* __restrict__ p, size_t n4) {
    size_t i      = (size_t)blockIdx.x * blockDim.x + threadIdx.x;
    size_t stride = (size_t)gridDim.x * blockDim.x;
    f32x4 z = {0.f, 0.f, 0.f, 0.f};
    for (; i < n4; i += stride)
        __builtin_nontemporal_store(z, p + i);
}

// ---------------------------------------------------------------------------
// 2) router: WMMA fp32 GEMM (16 tokens/wave), softmax, top-2, loss partials
// ---------------------------------------------------------------------------
__global__ __launch_bounds__(256)
void router_wmma_kernel(const float* __restrict__ X,    // [B*N, D]
                        const float* __restrict__ W,    // [D, E]
                        TokRec* __restrict__ tok,       // [B*N]
                        float*  __restrict__ accbuf)
{
    __shared__ float lw[D_ * 16];        // w_gate padded to 16 cols (64 KB)
    __shared__ float lg[8][16][8];       // per-wave logits: [wave][token][expert]

    const int tid = threadIdx.x;

    // stage w_gate into LDS, zero-padding columns 8..15 so WMMA N=16 is exact
    for (int i = tid; i < D_ * 16; i += 256) {
        int col = i & 15;
        lw[i] = (col < E_) ? W[(i >> 4) * E_ + col] : 0.0f;
    }
    __syncthreads();

    const int wave = tid >> 5;
    const int lane = tid & 31;
    const int tile = blockIdx.x * 8 + wave;      // 0..1023, 16 tokens each
    const int m    = lane & 15;                  // A: token row / B: column n
    const int hi   = lane >> 4;                  // K-half selector

    const float* arow = X + ((size_t)tile * 16 + m) * D_;

    f32x8 acc = {0.f, 0.f, 0.f, 0.f, 0.f, 0.f, 0.f, 0.f};
    for (int k0 = 0; k0 < D_; k0 += 4) {
        const int ka = k0 + 2 * hi;
        // A 16x4 f32 layout: v0 = K=ka, v1 = K=ka+1 (lanes 0-15: K 0/1, lanes 16-31: K 2/3)
        f32x2 a = *(const f32x2*)(arow + ka);
        // B 4x16 f32 layout: row K striped across lanes; same K split by lane half
        f32x2 b;
        b.x = lw[(ka    ) * 16 + m];
        b.y = lw[(ka + 1) * 16 + m];
        acc = __builtin_amdgcn_wmma_f32_16x16x4_f32(
                  /*neg_a=*/false, a, /*neg_b=*/false, b,
                  /*c_mod=*/(short)0, acc, /*reuse_a=*/false, /*reuse_b=*/false);
    }

    // C/D layout: VGPR r, lanes 0-15 -> (M=r, N=lane); lanes 16-31 -> (M=r+8, N=lane-16)
    if (m < E_) {
        #pragma unroll
        for (int r = 0; r < 8; ++r)
            lg[wave][r + 8 * hi][m] = acc[r];
    }
    __syncthreads();

    float pv[8], cv[8];
    float zz = 0.0f;
    #pragma unroll
    for (int e = 0; e < 8; ++e) { pv[e] = 0.0f; cv[e] = 0.0f; }

    if (lane < 16) {
        float l[8];
        #pragma unroll
        for (int e = 0; e < 8; ++e) l[e] = lg[wave][lane][e];

        float mx = l[0];
        #pragma unroll
        for (int e = 1; e < 8; ++e) mx = fmaxf(mx, l[e]);

        float s = 0.0f, ex[8];
        #pragma unroll
        for (int e = 0; e < 8; ++e) { ex[e] = __expf(l[e] - mx); s += ex[e]; }
        const float inv = 1.0f / s;
        #pragma unroll
        for (int e = 0; e < 8; ++e) pv[e] = ex[e] * inv;

        const float lse = mx + __logf(s);
        zz = lse * lse;

        // top-2 (strict > : ties resolve to lower index, matching lax.top_k)
        int e1 = 0; float p1 = pv[0];
        #pragma unroll
        for (int e = 1; e < 8; ++e) if (pv[e] > p1) { p1 = pv[e]; e1 = e; }
        int e2 = -1; float p2 = -1.0f;
        #pragma unroll
        for (int e = 0; e < 8; ++e) if (e != e1 && pv[e] > p2) { p2 = pv[e]; e2 = e; }

        cv[e1] = 1.0f;
        cv[e2] = 1.0f;

        const int gtok = tile * 16 + lane;
        tok[gtok] = TokRec{e1, e2, p1, p2};
    }

    // wave32 xor-shuffle reduction of loss partials (lanes >=16 contribute 0)
    #pragma unroll
    for (int e = 0; e < 8; ++e) {
        float v = pv[e], c = cv[e];
        for (int off = 16; off >= 1; off >>= 1) {
            v += __shfl_xor(v, off);
            c += __shfl_xor(c, off);
        }
        pv[e] = v; cv[e] = c;
    }
    for (int off = 16; off >= 1; off >>= 1) zz += __shfl_xor(zz, off);

    if (lane == 0) {
        const int b = (tile * 16) >> 12;   // token / 4096
        atomicAdd(&accbuf[0], zz);
        #pragma unroll
        for (int e = 0; e < 8; ++e) {
            atomicAdd(&accbuf[1  + b * 8 + e], pv[e]);
            atomicAdd(&accbuf[33 + b * 8 + e], cv[e]);
        }
    }
}

// ---------------------------------------------------------------------------
// 3) per-batch priority scan + sparse scatter of dispatch/combine nonzeros
//    stream order s = k*N + n (top-k-major), priority = # earlier picks of e
// ---------------------------------------------------------------------------
__global__ __launch_bounds__(1024)
void priority_kernel(const TokRec* __restrict__ tok, float* __restrict__ out)
{
    __shared__ int cnts[1024][8];     // 32 KB
    __shared__ int waveTot[32][8];
    __shared__ int waveOff[32][8];

    const int b   = blockIdx.x;
    const int tid = threadIdx.x;
    const TokRec* tb = tok + (size_t)b * N_;

    int   myE[8]; float myP[8]; int myN[8];
    int   c[8];
    #pragma unroll
    for (int e = 0; e < 8; ++e) c[e] = 0;

    #pragma unroll
    for (int j = 0; j < 8; ++j) {
        const int s = tid * 8 + j;          // 0..8191
        const int k = s >> 12;              // slot
        const int n = s & (N_ - 1);
        const TokRec r = tb[n];
        const int   e = k ? r.e2 : r.e1;
        const float p = k ? r.p2 : r.p1;
        myE[j] = e; myP[j] = p; myN[j] = n;
        c[e]++;
    }
    #pragma unroll
    for (int e = 0; e < 8; ++e) cnts[tid][e] = c[e];
    __syncthreads();

    // level 1: totals per wave of 32 threads
    if (tid < 256) {
        const int w = tid >> 3, e = tid & 7;
        int ssum = 0;
        for (int j = 0; j < 32; ++j) ssum += cnts[w * 32 + j][e];
        waveTot[w][e] = ssum;
    }
    __syncthreads();

    // level 2: exclusive scan over 32 wave totals (8 threads, one per expert)
    if (tid < 8) {
        int run = 0;
        for (int w = 0; w < 32; ++w) { waveOff[w][tid] = run; run += waveTot[w][tid]; }
    }
    __syncthreads();

    // per-thread exclusive base offsets
    int base[8];
    {
        const int w = tid >> 5, l = tid & 31;
        #pragma unroll
        for (int e = 0; e < 8; ++e) {
            int s = waveOff[w][e];
            for (int j = 0; j < l; ++j) s += cnts[w * 32 + j][e];
            base[e] = s;
        }
    }

    float* __restrict__ disp = out;
    float* __restrict__ comb = out + BNEC;

    #pragma unroll
    for (int j = 0; j < 8; ++j) {
        const int e = myE[j];
        const int p = base[e]++;
        if (p < CAP_) {
            const size_t off = (((size_t)b * N_ + myN[j]) * E_ + e) * CAP_ + p;
            disp[off] = 1.0f;
            comb[off] = myP[j];
        }
    }
}

// ---------------------------------------------------------------------------
// 4) scalars: aux_loss, z_loss
// ---------------------------------------------------------------------------
__global__ void finalize_kernel(const float* __restrict__ accbuf, float* __restrict__ out)
{
    if (threadIdx.x == 0 && blockIdx.x == 0) {
        float aux = 0.0f;
        for (int i = 0; i < B_ * E_; ++i) {
            const float cnt = accbuf[33 + i];
            const float ps  = accbuf[1 + i];
            aux += (cnt / (float)N_) * (ps / (float)N_);
        }
        aux = aux * (float)(E_ * E_) / (float)(B_ * E_);
        const float z = accbuf[0] / (float)(B_ * N_);
        out[2 * BNEC]     = aux;
        out[2 * BNEC + 1] = z;
    }
}

// ---------------------------------------------------------------------------
extern "C" void kernel_launch(void* const* d_in, const int* in_sizes, int n_in,
                              void* d_out, int out_size, void* d_ws, size_t ws_size,
                              hipStream_t stream) {
    (void)in_sizes; (void)n_in; (void)out_size; (void)ws_size;

    const float* X = (const float*)d_in[0];   // token_inputs [B,N,D] f32
    const float* W = (const float*)d_in[1];   // w_gate [D,E] f32
    // d_in[2] is expert_capacity (scalar int) == CAP_, compile-time constant here

    float*  out    = (float*)d_out;
    float*  accbuf = (float*)d_ws;                            // 65 floats used
    TokRec* tok    = (TokRec*)((char*)d_ws + 4096);           // [B*N] records

    init_ws_kernel<<<1, 256, 0, stream>>>(accbuf);

    const size_t n4 = (2 * BNEC) / 4;                         // float4 count
    zero_kernel<<<4096, 256, 0, stream>>>((f32x4*)out, n4);

    router_wmma_kernel<<<128, 256, 0, stream>>>(X, W, tok, accbuf);

    priority_kernel<<<B_, 1024, 0, stream>>>(tok, out);

    finalize_kernel<<<1, 64, 0, stream>>>(accbuf, out);
}